// LocalPredictiveAttention_73718818668970
// MI455X (gfx1250) — compile-verified
//
#include <hip/hip_runtime.h>
#include <cstddef>

typedef __attribute__((ext_vector_type(2))) float v2f;
typedef __attribute__((ext_vector_type(8))) float v8f;

#define S_LEN  2048
#define BATCH  64
#define HDIM   1024
#define DHDIM  1024
#define DHALF  64
#define WWIN   129   // 2*DHALF + 1

// ---------------------------------------------------------------------------
// Kernel A: fused predictive-projection GEMM.
//   C[b,h] = sum_k hidden[b,k] * Wp_w[h,k]   (64x1024x1024, fp32 WMMA 16x16x4)
// folded into part[b, htile] = sum_{h in tile} tanh(C + bias[h]) * vp_w[h].
// One wave per 16x16 tile; EXEC all-ones through the WMMA loop.
// ---------------------------------------------------------------------------
__global__ __launch_bounds__(32)
void predict_gemm_kernel(const float* __restrict__ hidden,
                         const float* __restrict__ Wp_w,
                         const float* __restrict__ Wp_b,
                         const float* __restrict__ vp_w,
                         float* __restrict__ part) {
  const int htile = blockIdx.x;       // 0..63  (H/16)
  const int btile = blockIdx.y;       // 0..3   (B/16)
  const int lane  = threadIdx.x;      // 0..31
  const int half  = lane >> 4;        // K sub-pair select (ISA 16x4 fp32 A layout)
  const int lr    = lane & 15;

  const int m = btile * 16 + lr;      // batch row (A matrix M)
  const int h = htile * 16 + lr;      // output column (B matrix N)

  // Lane holds K = {2*half, 2*half+1} of each 4-wide K step (VGPR0/VGPR1).
  const float* Arow = hidden + (size_t)m * DHDIM + 2 * half;
  const float* Brow = Wp_w   + (size_t)h * DHDIM + 2 * half;  // B[k][n] = Wp_w[h,k]

  v8f c = {};
  #pragma unroll 4
  for (int k0 = 0; k0 < DHDIM; k0 += 4) {
    v2f a = *(const v2f*)(Arow + k0);
    v2f b = *(const v2f*)(Brow + k0);
    // 8 args: (neg_a, A, neg_b, B, c_mod, C, reuse_a, reuse_b)
    c = __builtin_amdgcn_wmma_f32_16x16x4_f32(false, a, false, b,
                                              (short)0, c, false, false);
  }

  // C/D layout: VGPR r -> M = r + 8*half, N = lr fixed per lane.
  const float bias = Wp_b[h];
  const float vw   = vp_w[h];
  #pragma unroll
  for (int r = 0; r < 8; ++r) {
    float t = tanhf(c[r] + bias) * vw;
    t += __shfl_xor(t, 1);
    t += __shfl_xor(t, 2);
    t += __shfl_xor(t, 4);
    t += __shfl_xor(t, 8);          // butterfly within each 16-lane half
    if (lr == 0) {
      const int b = btile * 16 + r + 8 * half;
      part[b * 64 + htile] = t;     // written exactly once -> deterministic
    }
  }
}

// ---------------------------------------------------------------------------
// Kernel B: p[b] = S * sigmoid(sum_tiles part[b] + vp_b); center = rint(p).
// ---------------------------------------------------------------------------
__global__ __launch_bounds__(64)
void predict_pos_kernel(const float* __restrict__ part,
                        const float* __restrict__ vp_b,
                        float* __restrict__ pbuf,
                        int* __restrict__ cbuf) {
  const int b = threadIdx.x;
  float s = 0.f;
  #pragma unroll 8
  for (int j = 0; j < 64; ++j) s += part[b * 64 + j];
  s += vp_b[0];
  const float p = (float)S_LEN / (1.f + expf(-s));   // S * sigmoid(s)
  pbuf[b] = p;
  cbuf[b] = (int)rintf(p);          // round-half-even, matches jnp.round
}

// ---------------------------------------------------------------------------
// Kernel C1: window scores. One wave per (batch, window-row) pair:
//   score[b,w] = hidden[b,:] . enc[center[b]+w-D, b, :]   (0 if out of range)
// 8256 waves total; float4 (b128) coalesced loads + wave butterfly reduce.
// ---------------------------------------------------------------------------
__global__ __launch_bounds__(256)
void scores_kernel(const float* __restrict__ hidden,
                   const float* __restrict__ enc,
                   const int* __restrict__ cbuf,
                   float* __restrict__ scorebuf) {
  const int wave = threadIdx.x >> 5;
  const int lane = threadIdx.x & 31;
  const int pair = blockIdx.x * 8 + wave;      // exact grid: 1032*8 = 64*129
  const int b    = pair / WWIN;
  const int w    = pair - b * WWIN;
  const int s    = cbuf[b] + w - DHALF;

  float acc = 0.f;
  if (s >= 0 && s < S_LEN) {                   // wave-uniform branch
    const float4* hrow = (const float4*)(hidden + (size_t)b * HDIM);
    const float4* erow = (const float4*)(enc + (size_t)s * (BATCH * HDIM)
                                             + (size_t)b * HDIM);
    #pragma unroll
    for (int i = 0; i < 8; ++i) {              // 32 lanes * 8 * float4 = 1024
      const float4 hv = hrow[i * 32 + lane];
      const float4 ev = erow[i * 32 + lane];
      acc += hv.x * ev.x + hv.y * ev.y + hv.z * ev.z + hv.w * ev.w;
    }
  }
  acc += __shfl_xor(acc, 1);
  acc += __shfl_xor(acc, 2);
  acc += __shfl_xor(acc, 4);
  acc += __shfl_xor(acc, 8);
  acc += __shfl_xor(acc, 16);
  if (lane == 0) scorebuf[pair] = acc;
}

// ---------------------------------------------------------------------------
// Kernel C2: per-batch softmax over the 129-wide window, then Gaussian
// rescale with stddev = D/2 = 32 (coef 1/2048), using the real-valued p.
// ---------------------------------------------------------------------------
__global__ __launch_bounds__(256)
void softmax_kernel(const float* __restrict__ scorebuf,
                    const float* __restrict__ pbuf,
                    const int* __restrict__ cbuf,
                    float* __restrict__ attn_out,
                    float* __restrict__ attnbuf) {
  __shared__ float red[2];
  const int b    = blockIdx.x;
  const int tid  = threadIdx.x;
  const int wave = tid >> 5;
  const int lane = tid & 31;
  const float* score = scorebuf + b * WWIN;

  if (wave == 0) {
    float m = -3.402823466e38f;
    for (int w = lane; w < WWIN; w += 32) m = fmaxf(m, score[w]);
    m = fmaxf(m, __shfl_xor(m, 1));
    m = fmaxf(m, __shfl_xor(m, 2));
    m = fmaxf(m, __shfl_xor(m, 4));
    m = fmaxf(m, __shfl_xor(m, 8));
    m = fmaxf(m, __shfl_xor(m, 16));
    float sum = 0.f;
    for (int w = lane; w < WWIN; w += 32) sum += expf(score[w] - m);
    sum += __shfl_xor(sum, 1);
    sum += __shfl_xor(sum, 2);
    sum += __shfl_xor(sum, 4);
    sum += __shfl_xor(sum, 8);
    sum += __shfl_xor(sum, 16);
    if (lane == 0) { red[0] = m; red[1] = sum; }
  }
  __syncthreads();
  const float maxv = red[0];
  const float rsum = 1.f / red[1];

  if (tid < WWIN) {
    const float a    = expf(score[tid] - maxv) * rsum;
    const float widx = (float)(cbuf[b] + tid - DHALF);   // real position
    const float d    = widx - pbuf[b];
    const float as   = a * expf(-(1.0f / 2048.0f) * d * d);
    attn_out[b * WWIN + tid] = as;
    attnbuf[b * WWIN + tid]  = as;
  }
}

// ---------------------------------------------------------------------------
// Kernel C3: context[b,h] = sum_w attn[b,w] * enc_local[w, b, h].
// Grid (H/256, B): each block streams 129 coalesced 1KB row segments,
// prefetching the next window row speculatively.
// ---------------------------------------------------------------------------
__global__ __launch_bounds__(256)
void context_kernel(const float* __restrict__ enc,
                    const float* __restrict__ attnbuf,
                    const int* __restrict__ cbuf,
                    float* __restrict__ ctx_out) {
  __shared__ float attn[WWIN];
  const int b      = blockIdx.y;
  const int h      = blockIdx.x * 256 + threadIdx.x;
  const int center = cbuf[b];

  for (int i = threadIdx.x; i < WWIN; i += 256)
    attn[i] = attnbuf[b * WWIN + i];
  __syncthreads();

  const float* base = enc + (size_t)b * HDIM + h;
  float acc = 0.f;
  for (int w = 0; w < WWIN; ++w) {
    const int s = center + w - DHALF;
    if (s < 0 || s >= S_LEN) continue;         // zero padding
    const int sn = s + 1;
    if (sn < S_LEN)                            // speculative next-row prefetch
      __builtin_prefetch(base + (size_t)sn * (BATCH * HDIM), 0, 1);
    acc += attn[w] * base[(size_t)s * (BATCH * HDIM)];
  }
  ctx_out[(size_t)b * HDIM + h] = acc;
}

// ---------------------------------------------------------------------------
// Inputs (setup_inputs order): t, hidden[64,1024], encoder_outputs[2048,64,1024],
// Wp_w[1024,1024], Wp_b[1024], vp_w[1,1024], vp_b[1].
// Output: attn_scaled[64,129] ++ context[64,1024]  (fp32).
// ---------------------------------------------------------------------------
extern "C" void kernel_launch(void* const* d_in, const int* in_sizes, int n_in,
                              void* d_out, int out_size, void* d_ws, size_t ws_size,
                              hipStream_t stream) {
  const float* hidden = (const float*)d_in[1];
  const float* enc    = (const float*)d_in[2];
  const float* Wp_w   = (const float*)d_in[3];
  const float* Wp_b   = (const float*)d_in[4];
  const float* vp_w   = (const float*)d_in[5];
  const float* vp_b   = (const float*)d_in[6];

  float* out      = (float*)d_out;
  float* attn_out = out;                 // [64][129]
  float* ctx_out  = out + BATCH * WWIN;  // [64][1024]

  // Workspace layout (floats); every element rewritten each call.
  float* part     = (float*)d_ws;              // [64][64] GEMM tile partials
  float* pbuf     = part + BATCH * 64;         // [64] p
  int*   cbuf     = (int*)(pbuf + BATCH);      // [64] centers
  float* scorebuf = (float*)(cbuf + BATCH);    // [64][129] raw scores
  float* attnbuf  = scorebuf + BATCH * WWIN;   // [64][129] scaled attention

  dim3 gridA(HDIM / 16, BATCH / 16);           // 64 x 4 tiles, one wave each
  predict_gemm_kernel<<<gridA, 32, 0, stream>>>(hidden, Wp_w, Wp_b, vp_w, part);
  predict_pos_kernel<<<1, 64, 0, stream>>>(part, vp_b, pbuf, cbuf);

  scores_kernel<<<(BATCH * WWIN) / 8, 256, 0, stream>>>(hidden, enc, cbuf,
                                                        scorebuf);
  softmax_kernel<<<BATCH, 256, 0, stream>>>(scorebuf, pbuf, cbuf,
                                            attn_out, attnbuf);
  dim3 gridC3(HDIM / 256, BATCH);              // 4 x 64 blocks
  context_kernel<<<gridC3, 256, 0, stream>>>(enc, attnbuf, cbuf, ctx_out);
}